// IpaBlock_84782654423231
// MI455X (gfx1250) — compile-verified
//
#include <hip/hip_runtime.h>
#include <hip/hip_bf16.h>

// ---------------------------------------------------------------------------
// IPA-style graph attention block for MI455X (gfx1250, wave32, WMMA).
// Dense 128x128 projections use V_WMMA_F32_16X16X4_F32 (fp32 matrix pipe,
// matches reference precision; GEMMs are bandwidth-bound anyway).
// Edge phase is L2-resident (q/k/v = 76.8MB << 192MB L2) -> atomic multi-pass.
// ---------------------------------------------------------------------------

#define N_NODES 50000
#define N_EDGES 500000
#define HIDDEN  128
#define HEADS   8
#define DH      16

typedef float wvec2 __attribute__((ext_vector_type(2)));
typedef float wvec8 __attribute__((ext_vector_type(8)));

// Workspace layout (float offsets)
#define OFF_Q    0u
#define OFF_K    6400000u
#define OFF_V    12800000u
#define OFF_LG   19200000u   // E*8, logits then overwritten by exp()
#define OFF_M    23200000u   // N*8 segment max
#define OFF_S    23600000u   // N*8 segment sum      (zeroed)
#define OFF_HUPD 24000000u   // N*128 h update       (zeroed)
#define OFF_DISP 30400000u   // N*3 displacement     (zeroed)
#define OFF_T    30550000u   // N*128 silu hidden

// ---------------------------------------------------------------------------
// fp32 GEMM: out[N,128] = act( A[N,128] @ W[128,128] + bias (+ addsrc) )
// One wave -> 16x16 tile, 8 waves/block cover all 128 output cols.
// grid.x = N/16 (50000/16 = 3125 exact), block = 256.
// ---------------------------------------------------------------------------
__global__ void gemm128_wmma(const float* __restrict__ A,
                             const float* __restrict__ W,
                             const float* __restrict__ bias,
                             const float* __restrict__ addsrc,   // may be null
                             float* __restrict__ out,
                             int act)                            // 0=none 1=silu
{
    const int wave = threadIdx.x >> 5;
    const int lane = threadIdx.x & 31;
    const int m0   = blockIdx.x * 16;
    const int n0   = wave * 16;
    const int row  = m0 + (lane & 15);
    const int col  = n0 + (lane & 15);
    const int kh   = (lane >> 4) << 1;      // 0 or 2: which K pair this half-wave holds

    const float* __restrict__ arow = A + row * HIDDEN;
    wvec8 acc = {};

#pragma unroll
    for (int k = 0; k < HIDDEN; k += 4) {
        // A 16x4 tile: lane M = lane&15, VGPR0/1 = K = k+kh, k+kh+1
        wvec2 a = *(const wvec2*)(arow + k + kh);
        // B 4x16 tile: lane N = lane&15, VGPR0/1 = K rows k+kh, k+kh+1
        wvec2 b;
        b.x = W[(k + kh)     * HIDDEN + col];
        b.y = W[(k + kh + 1) * HIDDEN + col];
        acc = __builtin_amdgcn_wmma_f32_16x16x4_f32(
                  false, a, false, b, (short)0, acc, false, false);
    }

    const float bcol  = bias[col];
    const int   mbase = m0 + ((lane >> 4) << 3);   // D: vgpr r -> M = r + 8*lane[4]
#pragma unroll
    for (int r = 0; r < 8; ++r) {
        const int m = mbase + r;
        float v = acc[r] + bcol;
        if (addsrc) v += addsrc[m * HIDDEN + col];
        if (act == 1) v = v * (1.0f / (1.0f + __expf(-v)));   // SiLU
        out[m * HIDDEN + col] = v;
    }
}

// ---------------------------------------------------------------------------
// init: fill segment-max with -inf
// ---------------------------------------------------------------------------
__global__ void fill_neginf(float* __restrict__ p, int n)
{
    int i = blockIdx.x * blockDim.x + threadIdx.x;
    if (i < n) p[i] = -__builtin_inff();
}

// float atomic max via signed/unsigned integer ordering trick
__device__ inline void atomicMaxF32(float* addr, float val)
{
    if (!(val < 0.0f)) {
        atomicMax((int*)addr, __float_as_int(val));
    } else {
        atomicMin((unsigned int*)addr, __float_as_uint(val));
    }
}

// ---------------------------------------------------------------------------
// edge logits: lg[e,h] = q[src]·k[dst]/sqrt(dh) - (d^2*Wd[h]+bd[h]); atomic max
// one thread per (edge, head). grid = E*8/256 = 15625.
// ---------------------------------------------------------------------------
__global__ void edge_logits(const float* __restrict__ q,
                            const float* __restrict__ k,
                            const int* __restrict__ src,
                            const int* __restrict__ dst,
                            const float* __restrict__ distances,
                            const float* __restrict__ Wd,
                            const float* __restrict__ bd,
                            float* __restrict__ lg,
                            float* __restrict__ m)
{
    int t = blockIdx.x * blockDim.x + threadIdx.x;
    if (t >= N_EDGES * HEADS) return;
    const int e  = t >> 3;
    const int hh = t & 7;
    const int sN = src[e];
    const int dN = dst[e];

    const float4* qp = (const float4*)(q + (size_t)sN * HIDDEN + hh * DH);
    const float4* kp = (const float4*)(k + (size_t)dN * HIDDEN + hh * DH);
    float dot = 0.0f;
#pragma unroll
    for (int i = 0; i < 4; ++i) {
        float4 qa = qp[i], kb = kp[i];
        dot += qa.x * kb.x + qa.y * kb.y + qa.z * kb.z + qa.w * kb.w;
    }
    const float d    = distances[e];
    const float bias = d * d * Wd[hh] + bd[hh];
    const float v    = dot * 0.25f - bias;        // 1/sqrt(16)
    lg[(size_t)e * HEADS + hh] = v;
    atomicMaxF32(&m[(size_t)sN * HEADS + hh], v);
}

// ---------------------------------------------------------------------------
// ex = exp(lg - m[src]); atomic-add segment sum. Overwrites lg with ex.
// ---------------------------------------------------------------------------
__global__ void edge_exp(float* __restrict__ lg,
                         const float* __restrict__ m,
                         const int* __restrict__ src,
                         float* __restrict__ s)
{
    int t = blockIdx.x * blockDim.x + threadIdx.x;
    if (t >= N_EDGES * HEADS) return;
    const int e  = t >> 3;
    const int hh = t & 7;
    const int sN = src[e];
    const float ex = __expf(lg[(size_t)e * HEADS + hh] - m[(size_t)sN * HEADS + hh]);
    lg[(size_t)e * HEADS + hh] = ex;
    atomicAdd(&s[(size_t)sN * HEADS + hh], ex);
}

// ---------------------------------------------------------------------------
// scatter: h_upd[src] += v[dst]*w ; disp[src] += (x[dst]-x[src])*mean_h(w)
// one thread per (edge, dim). grid = E*128/256 = 250000.
// ---------------------------------------------------------------------------
__global__ void edge_scatter(const float* __restrict__ ex,
                             const float* __restrict__ s,
                             const float* __restrict__ v,
                             const float* __restrict__ x,
                             const int* __restrict__ src,
                             const int* __restrict__ dst,
                             float* __restrict__ h_upd,
                             float* __restrict__ disp)
{
    long long t = (long long)blockIdx.x * blockDim.x + threadIdx.x;
    const int e   = (int)(t >> 7);
    const int dim = (int)(t & 127);
    const int hh  = dim >> 4;
    const int sN  = src[e];
    const int dN  = dst[e];

    const float w = ex[(size_t)e * HEADS + hh] /
                    fmaxf(s[(size_t)sN * HEADS + hh], 1e-9f);
    atomicAdd(&h_upd[(size_t)sN * HIDDEN + dim],
              v[(size_t)dN * HIDDEN + dim] * w);

    if (dim < 3) {
        float wm = 0.0f;
#pragma unroll
        for (int h2 = 0; h2 < HEADS; ++h2)
            wm += ex[(size_t)e * HEADS + h2] /
                  fmaxf(s[(size_t)sN * HEADS + h2], 1e-9f);
        wm *= (1.0f / HEADS);
        atomicAdd(&disp[(size_t)sN * 3 + dim],
                  (x[(size_t)dN * 3 + dim] - x[(size_t)sN * 3 + dim]) * wm);
    }
}

// ---------------------------------------------------------------------------
// gate = tanh(t @ Wg2 + bg2); x_out = x + gate*disp. One wave32 per node.
// grid = N/8 = 6250 blocks of 256.
// ---------------------------------------------------------------------------
__global__ void gate_xout(const float* __restrict__ tbuf,
                          const float* __restrict__ Wg2,
                          const float* __restrict__ bg2,
                          const float* __restrict__ x,
                          const float* __restrict__ disp,
                          float* __restrict__ xout)
{
    const int wave = threadIdx.x >> 5;
    const int lane = threadIdx.x & 31;
    const int n    = blockIdx.x * 8 + wave;
    if (n >= N_NODES) return;

    float sum = 0.0f;
#pragma unroll
    for (int i = lane; i < HIDDEN; i += 32)
        sum += tbuf[(size_t)n * HIDDEN + i] * Wg2[i];
#pragma unroll
    for (int off = 16; off >= 1; off >>= 1)
        sum += __shfl_xor(sum, off, 32);

    const float gate = tanhf(sum + bg2[0]);
    if (lane < 3)
        xout[(size_t)n * 3 + lane] = x[(size_t)n * 3 + lane] +
                                     gate * disp[(size_t)n * 3 + lane];
}

// ---------------------------------------------------------------------------
extern "C" void kernel_launch(void* const* d_in, const int* in_sizes, int n_in,
                              void* d_out, int out_size, void* d_ws, size_t ws_size,
                              hipStream_t stream)
{
    const float* h    = (const float*)d_in[0];
    const float* x    = (const float*)d_in[1];
    const int*   src  = (const int*)d_in[2];
    const int*   dst  = (const int*)d_in[3];
    const float* dist = (const float*)d_in[4];
    const float* Wq = (const float*)d_in[5];   const float* bq = (const float*)d_in[6];
    const float* Wk = (const float*)d_in[7];   const float* bk = (const float*)d_in[8];
    const float* Wv = (const float*)d_in[9];   const float* bv = (const float*)d_in[10];
    const float* Wo = (const float*)d_in[11];  const float* bo = (const float*)d_in[12];
    const float* Wd = (const float*)d_in[13];  const float* bd = (const float*)d_in[14];
    const float* Wg1 = (const float*)d_in[15]; const float* bg1 = (const float*)d_in[16];
    const float* Wg2 = (const float*)d_in[17]; const float* bg2 = (const float*)d_in[18];

    float* ws    = (float*)d_ws;
    float* q     = ws + OFF_Q;
    float* kk    = ws + OFF_K;
    float* vv    = ws + OFF_V;
    float* lg    = ws + OFF_LG;
    float* mbuf  = ws + OFF_M;
    float* sbuf  = ws + OFF_S;
    float* h_upd = ws + OFF_HUPD;
    float* disp  = ws + OFF_DISP;
    float* tbuf  = ws + OFF_T;

    float* h_out = (float*)d_out;                        // [N,128]
    float* x_out = (float*)d_out + (size_t)N_NODES * HIDDEN; // [N,3]

    // init: zeros for s + h_upd + disp (contiguous), -inf for m
    hipMemsetAsync(sbuf, 0,
                   (size_t)(OFF_T - OFF_S) * sizeof(float), stream);
    fill_neginf<<<(N_NODES * HEADS + 255) / 256, 256, 0, stream>>>(
        mbuf, N_NODES * HEADS);

    // Q/K/V projections (WMMA fp32)
    dim3 blk(256);
    dim3 gGemm(N_NODES / 16);   // 3125
    gemm128_wmma<<<gGemm, blk, 0, stream>>>(h, Wq, bq, nullptr, q, 0);
    gemm128_wmma<<<gGemm, blk, 0, stream>>>(h, Wk, bk, nullptr, kk, 0);
    gemm128_wmma<<<gGemm, blk, 0, stream>>>(h, Wv, bv, nullptr, vv, 0);

    // edge attention (L2-resident gathers + atomics)
    edge_logits<<<(N_EDGES * HEADS) / 256, blk, 0, stream>>>(
        q, kk, src, dst, dist, Wd, bd, lg, mbuf);
    edge_exp<<<(N_EDGES * HEADS) / 256, blk, 0, stream>>>(lg, mbuf, src, sbuf);
    edge_scatter<<<(N_EDGES * HIDDEN) / 256, blk, 0, stream>>>(
        lg, sbuf, vv, x, src, dst, h_upd, disp);

    // h_out = h + h_upd @ Wo + bo  (WMMA, written straight to d_out)
    gemm128_wmma<<<gGemm, blk, 0, stream>>>(h_upd, Wo, bo, h, h_out, 0);
    // tbuf = silu(h_out @ Wg1 + bg1)
    gemm128_wmma<<<gGemm, blk, 0, stream>>>(h_out, Wg1, bg1, nullptr, tbuf, 1);
    // gate + x_out
    gate_xout<<<N_NODES / 8, blk, 0, stream>>>(tbuf, Wg2, bg2, x, disp, x_out);
}